// GATv2Net_NodeClassifier_67370857005645
// MI455X (gfx1250) — compile-verified
//
#include <hip/hip_runtime.h>
#include <hip/hip_bf16.h>
#include <math.h>

#define N_NODES 50000
#define E_EDGES 800000
#define NUM_CLASSES 10
#define OUT_C 64
#define NEG_SLOPE 0.2f
#define GK 128            // K is 128 for both GATv2 layer GEMMs
#define LDS_PITCH 132     // 128 + 4 floats pad -> conflict-free ds_load_b128

typedef __attribute__((ext_vector_type(2))) float v2f;
typedef __attribute__((ext_vector_type(8))) float v8f;

__device__ __forceinline__ unsigned encodeF(float x) {
  unsigned u = __float_as_uint(x);
  return (u & 0x80000000u) ? ~u : (u | 0x80000000u);
}
__device__ __forceinline__ float decodeF(unsigned u) {
  unsigned b = (u & 0x80000000u) ? (u & 0x7FFFFFFFu) : ~u;
  return __uint_as_float(b);
}

// ---------------------------------------------------------------------------
// GEMM: out[m,n] = sum_k X[m,k]*W[n,k] + bias[n],  K fixed = 128.
// Block = 8 waves. Block computes a 128(M) x 64(N) tile:
//   - W tile (64 rows x 128 K) staged cooperatively into padded LDS once.
//   - Each wave owns one 16x64 tile: preloads its full A slice as 16
//     global_load_b128 (all in flight at once), then runs 16 k-groups of
//     (4 x ds_load_b128 from LDS  +  8 x V_WMMA_F32_16X16X4_F32).
// k-permutation: a contiguous float4 per lane feeds two WMMA steps:
//   group g covers global k in [8g, 8g+8): step0 slots = {8g,8g+1 | 8g+4,8g+5},
//   step1 slots = {8g+2,8g+3 | 8g+6,8g+7}; identical assignment for A and B.
// ---------------------------------------------------------------------------
__global__ __launch_bounds__(256) void gemm_wmma(
    const float* __restrict__ X, const float* __restrict__ W,
    const float* __restrict__ bias, float* __restrict__ out,
    int M, int NC) {
  __shared__ float lw[64 * LDS_PITCH];

  int ntn = NC >> 6;
  int mg = blockIdx.x / ntn;        // group of 8 M-tiles
  int tn = blockIdx.x % ntn;        // 64-wide N tile
  int tid = threadIdx.x;
  int wv  = tid >> 5;
  int lane = tid & 31;
  int l15 = lane & 15;
  int hi  = lane >> 4;

  // ---- stage W[tn*64 .. tn*64+63][0..127] into LDS (float4 granularity) ----
  {
    const float4* wsrc = (const float4*)(W + (size_t)tn * 64 * GK);
    // 64 rows x 32 float4 = 2048 float4s; 256 threads x 8
#pragma unroll
    for (int i = 0; i < 8; ++i) {
      int idx = tid + i * 256;
      int row = idx >> 5;
      int kq  = idx & 31;
      float4 v = wsrc[row * 32 + kq];
      float* dst = &lw[row * LDS_PITCH + kq * 4];
      dst[0] = v.x; dst[1] = v.y; dst[2] = v.z; dst[3] = v.w;
    }
  }
  __syncthreads();

  int tm = mg * 8 + wv;
  if (tm < (M >> 4)) {              // wave-uniform guard: EXEC stays all-1s
    // ---- preload full A slice: 16 independent b128 loads ----
    const float* xrow = X + (size_t)(tm * 16 + l15) * GK;
    float4 areg[16];
#pragma unroll
    for (int g = 0; g < 16; ++g)
      areg[g] = *(const float4*)(xrow + g * 8 + hi * 4);

    v8f acc0 = {}, acc1 = {}, acc2 = {}, acc3 = {};
    const float* lb0 = &lw[(0 * 16 + l15) * LDS_PITCH + hi * 4];
    const float* lb1 = &lw[(1 * 16 + l15) * LDS_PITCH + hi * 4];
    const float* lb2 = &lw[(2 * 16 + l15) * LDS_PITCH + hi * 4];
    const float* lb3 = &lw[(3 * 16 + l15) * LDS_PITCH + hi * 4];

#pragma unroll
    for (int g = 0; g < 16; ++g) {
      float4 a = areg[g];
      v2f aA = { a.x, a.y };
      v2f aB = { a.z, a.w };
      float4 b0 = *(const float4*)(lb0 + g * 8);
      float4 b1 = *(const float4*)(lb1 + g * 8);
      float4 b2 = *(const float4*)(lb2 + g * 8);
      float4 b3 = *(const float4*)(lb3 + g * 8);
      v2f b; 
      b = (v2f){ b0.x, b0.y };
      acc0 = __builtin_amdgcn_wmma_f32_16x16x4_f32(false, aA, false, b, (short)0, acc0, false, false);
      b = (v2f){ b1.x, b1.y };
      acc1 = __builtin_amdgcn_wmma_f32_16x16x4_f32(false, aA, false, b, (short)0, acc1, false, false);
      b = (v2f){ b2.x, b2.y };
      acc2 = __builtin_amdgcn_wmma_f32_16x16x4_f32(false, aA, false, b, (short)0, acc2, false, false);
      b = (v2f){ b3.x, b3.y };
      acc3 = __builtin_amdgcn_wmma_f32_16x16x4_f32(false, aA, false, b, (short)0, acc3, false, false);
      b = (v2f){ b0.z, b0.w };
      acc0 = __builtin_amdgcn_wmma_f32_16x16x4_f32(false, aB, false, b, (short)0, acc0, false, false);
      b = (v2f){ b1.z, b1.w };
      acc1 = __builtin_amdgcn_wmma_f32_16x16x4_f32(false, aB, false, b, (short)0, acc1, false, false);
      b = (v2f){ b2.z, b2.w };
      acc2 = __builtin_amdgcn_wmma_f32_16x16x4_f32(false, aB, false, b, (short)0, acc2, false, false);
      b = (v2f){ b3.z, b3.w };
      acc3 = __builtin_amdgcn_wmma_f32_16x16x4_f32(false, aB, false, b, (short)0, acc3, false, false);
    }

    // C/D layout: VGPR v -> row v (+8 for lanes>=16), col = lane&15 per subtile
    int nbase = tn * 64 + l15;
    int rowbase = tm * 16 + hi * 8;
    float bb0 = bias[nbase], bb1 = bias[nbase + 16];
    float bb2 = bias[nbase + 32], bb3 = bias[nbase + 48];
#pragma unroll
    for (int v = 0; v < 8; ++v) {
      float* orow = out + (size_t)(rowbase + v) * NC + tn * 64 + l15;
      orow[0]  = acc0[v] + bb0;
      orow[16] = acc1[v] + bb1;
      orow[32] = acc2[v] + bb2;
      orow[48] = acc3[v] + bb3;
    }
  }
}

// ---------------------------------------------------------------------------
// Edge pass 1: score[e,h] = dot(leakyrelu(xl[src]+xr[dst]), att[h]);
//              atomic segment-max into mEnc[dst,h] (ordered-uint encoding).
// ---------------------------------------------------------------------------
__global__ __launch_bounds__(256) void edge_score(
    const float* __restrict__ xl, const float* __restrict__ xr,
    const float* __restrict__ att, const int* __restrict__ ei,
    float* __restrict__ score, unsigned* __restrict__ mEnc,
    int H, int C, int nE, int nTot) {
  int wave = blockIdx.x * (blockDim.x >> 5) + (threadIdx.x >> 5);
  int lane = threadIdx.x & 31;
  if (wave >= nTot) return;
  int s, d;
  if (wave < nE) { s = ei[wave]; d = ei[nE + wave]; }
  else { s = wave - nE; d = s; }               // appended self-loop
  const float* xs = xl + (size_t)s * H * C;
  const float* xd = xr + (size_t)d * H * C;
  for (int h = 0; h < H; ++h) {
    float part = 0.f;
    for (int c = lane; c < C; c += 32) {
      float e = xs[h * C + c] + xd[h * C + c];
      e = e > 0.f ? e : NEG_SLOPE * e;
      part += e * att[h * C + c];
    }
    for (int off = 16; off > 0; off >>= 1) part += __shfl_xor(part, off, 32);
    if (lane == 0) {
      score[(size_t)wave * H + h] = part;
      atomicMax(&mEnc[(size_t)d * H + h], encodeF(part));
    }
  }
}

// ---------------------------------------------------------------------------
// Edge pass 2: ex = exp(score - m[dst]); in-place; atomicAdd denom[dst,h]
// ---------------------------------------------------------------------------
__global__ __launch_bounds__(256) void edge_exp(
    const int* __restrict__ ei, float* __restrict__ score,
    const unsigned* __restrict__ mEnc, float* __restrict__ denom,
    int H, int nE, int nTot) {
  int idx = blockIdx.x * blockDim.x + threadIdx.x;
  if (idx >= nTot * H) return;
  int e = idx / H, h = idx - e * H;
  int d = (e < nE) ? ei[nE + e] : (e - nE);
  float m = decodeF(mEnc[(size_t)d * H + h]);
  float ex = expf(score[idx] - m);
  score[idx] = ex;
  atomicAdd(&denom[(size_t)d * H + h], ex);
}

// ---------------------------------------------------------------------------
// Edge pass 3: out[dst,h,:] += (ex/denom[dst,h]) * xl[src,h,:]  (f32 atomics)
// ---------------------------------------------------------------------------
__global__ __launch_bounds__(256) void edge_agg(
    const float* __restrict__ xl, const float* __restrict__ ex,
    const float* __restrict__ denom, const int* __restrict__ ei,
    float* __restrict__ out, int H, int C, int nE, int nTot) {
  int wave = blockIdx.x * (blockDim.x >> 5) + (threadIdx.x >> 5);
  int lane = threadIdx.x & 31;
  if (wave >= nTot) return;
  int s, d;
  if (wave < nE) { s = ei[wave]; d = ei[nE + wave]; }
  else { s = wave - nE; d = s; }
  for (int h = 0; h < H; ++h) {
    float alpha = ex[(size_t)wave * H + h] / denom[(size_t)d * H + h];
    const float* xs = xl + (size_t)s * H * C + h * C;
    float* od = out + (size_t)d * H * C + h * C;
    for (int c = lane; c < C; c += 32)
      atomicAdd(&od[c], alpha * xs[c]);
  }
}

__global__ __launch_bounds__(256) void fill_f32(float* __restrict__ p, float v, int n) {
  int i = blockIdx.x * blockDim.x + threadIdx.x;
  if (i < n) p[i] = v;
}
__global__ __launch_bounds__(256) void fill_u32(unsigned* __restrict__ p, unsigned v, int n) {
  int i = blockIdx.x * blockDim.x + threadIdx.x;
  if (i < n) p[i] = v;
}
__global__ __launch_bounds__(256) void bias_elu(float* __restrict__ h,
                                                const float* __restrict__ bias,
                                                int NC, int n) {
  int i = blockIdx.x * blockDim.x + threadIdx.x;
  if (i < n) {
    float x = h[i] + bias[i % NC];
    h[i] = x > 0.f ? x : (expf(x) - 1.f);
  }
}
__global__ __launch_bounds__(256) void head_gemm(
    const float* __restrict__ h, const float* __restrict__ Wc,
    const float* __restrict__ bc, float* __restrict__ out, int M) {
  int idx = blockIdx.x * blockDim.x + threadIdx.x;
  if (idx >= M * NUM_CLASSES) return;
  int n = idx / NUM_CLASSES, cl = idx - n * NUM_CLASSES;
  const float* hr = h + (size_t)n * OUT_C;
  const float* wr = Wc + (size_t)cl * OUT_C;
  float acc = bc[cl];
#pragma unroll 8
  for (int c = 0; c < OUT_C; ++c) acc += hr[c] * wr[c];
  out[idx] = acc;
}

static inline int cdiv(long long a, long long b) { return (int)((a + b - 1) / b); }

extern "C" void kernel_launch(void* const* d_in, const int* in_sizes, int n_in,
                              void* d_out, int out_size, void* d_ws, size_t ws_size,
                              hipStream_t stream) {
  (void)in_sizes; (void)n_in; (void)out_size; (void)ws_size;
  const float* x     = (const float*)d_in[0];
  const int*   ei    = (const int*)d_in[1];
  const float* Wl1   = (const float*)d_in[3];
  const float* bl1   = (const float*)d_in[4];
  const float* Wr1   = (const float*)d_in[5];
  const float* br1   = (const float*)d_in[6];
  const float* att1  = (const float*)d_in[7];
  const float* bias1 = (const float*)d_in[8];
  const float* Wl2   = (const float*)d_in[9];
  const float* bl2   = (const float*)d_in[10];
  const float* Wr2   = (const float*)d_in[11];
  const float* br2   = (const float*)d_in[12];
  const float* att2  = (const float*)d_in[13];
  const float* bias2 = (const float*)d_in[14];
  const float* Wc    = (const float*)d_in[15];
  const float* bc    = (const float*)d_in[16];
  float* out = (float*)d_out;

  const int Nn = N_NODES, Ee = E_EDGES, Etot = Ee + Nn;

  // workspace layout (reused between layers)
  float*    A  = (float*)d_ws;                       // xl1 [N,128]; later xl2[N,64] + h2[N,64]
  float*    B  = A + (size_t)Nn * 128;               // xr1 [N,128]; later xr2[N,64]
  float*    Cb = B + (size_t)Nn * 128;               // h1 accumulator [N,128]
  float*    S  = Cb + (size_t)Nn * 128;              // per-edge scores/ex [Etot,2]
  unsigned* Mx = (unsigned*)(S + (size_t)Etot * 2);  // encoded segment max [N,2]
  float*    Dn = (float*)(Mx + (size_t)Nn * 2);      // softmax denom [N,2]
  float*    H2 = A + (size_t)Nn * 64;                // layer-2 output accumulator

  const int BLK = 256, WPB = BLK / 32;
  const int mgroups = cdiv(Nn / 16, 8);              // 8 M-tiles per block

  // ---------------- Layer 1 (H=2, C=64, concat) ----------------
  gemm_wmma<<<mgroups * 2, BLK, 0, stream>>>(x, Wl1, bl1, A, Nn, 128);
  gemm_wmma<<<mgroups * 2, BLK, 0, stream>>>(x, Wr1, br1, B, Nn, 128);
  fill_u32<<<cdiv(Nn * 2, BLK), BLK, 0, stream>>>(Mx, 0u, Nn * 2);
  fill_f32<<<cdiv(Nn * 2, BLK), BLK, 0, stream>>>(Dn, 0.f, Nn * 2);
  fill_f32<<<cdiv((long long)Nn * 128, BLK), BLK, 0, stream>>>(Cb, 0.f, Nn * 128);
  edge_score<<<cdiv(Etot, WPB), BLK, 0, stream>>>(A, B, att1, ei, S, Mx, 2, 64, Ee, Etot);
  edge_exp<<<cdiv((long long)Etot * 2, BLK), BLK, 0, stream>>>(ei, S, Mx, Dn, 2, Ee, Etot);
  edge_agg<<<cdiv(Etot, WPB), BLK, 0, stream>>>(A, S, Dn, ei, Cb, 2, 64, Ee, Etot);
  bias_elu<<<cdiv((long long)Nn * 128, BLK), BLK, 0, stream>>>(Cb, bias1, 128, Nn * 128);

  // ---------------- Layer 2 (H=1, C=64, mean==identity) ----------------
  gemm_wmma<<<mgroups * 1, BLK, 0, stream>>>(Cb, Wl2, bl2, A, Nn, 64);
  gemm_wmma<<<mgroups * 1, BLK, 0, stream>>>(Cb, Wr2, br2, B, Nn, 64);
  fill_u32<<<cdiv(Nn, BLK), BLK, 0, stream>>>(Mx, 0u, Nn);
  fill_f32<<<cdiv(Nn, BLK), BLK, 0, stream>>>(Dn, 0.f, Nn);
  fill_f32<<<cdiv((long long)Nn * 64, BLK), BLK, 0, stream>>>(H2, 0.f, Nn * 64);
  edge_score<<<cdiv(Etot, WPB), BLK, 0, stream>>>(A, B, att2, ei, S, Mx, 1, 64, Ee, Etot);
  edge_exp<<<cdiv(Etot, BLK), BLK, 0, stream>>>(ei, S, Mx, Dn, 1, Ee, Etot);
  edge_agg<<<cdiv(Etot, WPB), BLK, 0, stream>>>(A, S, Dn, ei, H2, 1, 64, Ee, Etot);
  bias_elu<<<cdiv((long long)Nn * 64, BLK), BLK, 0, stream>>>(H2, bias2, 64, Nn * 64);

  // ---------------- Classifier head ----------------
  head_gemm<<<cdiv((long long)Nn * NUM_CLASSES, BLK), BLK, 0, stream>>>(H2, Wc, bc, out, Nn);
}